// YOLO_loss_2241972928635
// MI455X (gfx1250) — compile-verified
//
#include <hip/hip_runtime.h>
#include <stdint.h>

// ---------------- problem constants (match setup_inputs) ----------------
#define HH 128
#define WW 128
#define AA 5
#define NCLS 5
#define CCH 50          // C = A*(5+NUM_CLASSES)
#define RESV 8.0f

// Low 32 bits of a generic pointer to __shared__ == LDS byte offset
// (flat LDS aperture keeps the wave-relative offset in addr[31:0]).
__device__ __forceinline__ unsigned lds_byte_off(const void* p) {
    return (unsigned)(unsigned long long)p;
}

// ---------------- kernel 1: clear the anchor map (-1 = no object) -------
__global__ void k_init(int* __restrict__ amap, int n) {
    int i = blockIdx.x * blockDim.x + threadIdx.x;
    if (i < n) amap[i] = -1;
}

// ---------------- kernel 2: scatter labels -> dense per-cell map --------
// Replicates _transform_train including the quirky anchor IoU.
__global__ void k_scatter(const float* __restrict__ labels,
                          const float* __restrict__ anchors, int N,
                          int* __restrict__ amap, float4* __restrict__ bmap,
                          float* __restrict__ cmap) {
    int n = blockIdx.x * blockDim.x + threadIdx.x;
    if (n >= N) return;
    float cx = labels[n * 6 + 0], cy = labels[n * 6 + 1];
    float bw = labels[n * 6 + 2], bh = labels[n * 6 + 3];
    float cls = labels[n * 6 + 4];
    int img = (int)labels[n * 6 + 5];
    int xc = (int)floorf(cx / RESV);
    int yc = (int)floorf(cy / RESV);

    float tl1x = cx - bw * 0.5f, tl1y = cy - bh * 0.5f;
    float br1x = cx + bw * 0.5f, br1y = cy + bh * 0.5f;
    float a1 = fmaxf(br1x - tl1x, 0.f) * fmaxf(br1y - tl1y, 0.f);
    int best = 0; float bestv = -1e30f;
    for (int a = 0; a < AA; ++a) {
        float aw = anchors[a * 2 + 0], ah = anchors[a * 2 + 1];
        // reference quirk: (xy - a/2, xy + a/2) re-read as (cx,cy,w,h)
        float c2x = cx - aw * 0.5f, c2y = cy - ah * 0.5f;
        float w2x = cx + aw * 0.5f, w2y = cy + ah * 0.5f;
        float tl2x = c2x - w2x * 0.5f, tl2y = c2y - w2y * 0.5f;
        float br2x = c2x + w2x * 0.5f, br2y = c2y + w2y * 0.5f;
        float sx = fmaxf(fminf(br1x, br2x) - fmaxf(tl1x, tl2x), 0.f);
        float sy = fmaxf(fminf(br1y, br2y) - fmaxf(tl1y, tl2y), 0.f);
        float inter = sx * sy;
        float a2 = fmaxf(br2x - tl2x, 0.f) * fmaxf(br2y - tl2y, 0.f);
        float iou = inter / (a1 + a2 - inter);
        if (iou > bestv) { bestv = iou; best = a; }   // argmax keeps first on ties
    }
    if (img < 0 || xc < 0 || yc < 0 || xc >= WW || yc >= HH) return;
    int idx = (img * HH + yc) * WW + xc;
    amap[idx] = best;
    bmap[idx] = make_float4(cx / RESV, cy / RESV, bw / RESV, bh / RESV); // box/res
    cmap[idx] = cls;
}

// ---------------- kernel 3: main loss pass ------------------------------
// One block per (b, y) row; 128 threads = 4 wave32. The 50 channel rows
// (512 B each = 32 lanes x 16 B) are split across the 4 waves; each wave
// issues ~13 CDNA5 async b128 global->LDS loads, waits on its own
// ASYNCcnt, then one barrier publishes the 25.6 KB tile to the block.
__global__ void __launch_bounds__(128)
k_main(const float* __restrict__ out, const float* __restrict__ anchors,
       const int* __restrict__ amap, const float4* __restrict__ bmap,
       const float* __restrict__ cmap, float* __restrict__ partials) {
    __shared__ __align__(16) float smem[CCH * WW];   // 25.6 KB staging tile
    __shared__ float red[4 * 8];

    const int bid = blockIdx.x;        // = b*H + y
    const int b = bid / HH;
    const int y = bid - b * HH;
    const int x = threadIdx.x;
    const int lane = threadIdx.x & 31, wid = threadIdx.x >> 5;

    // ---- async stage: out[b, c, y, :] row -> smem[c*W .. c*W+127] ----
    const float* row0 = out + (size_t)b * (CCH * HH * WW) + (size_t)y * WW;
    const unsigned lbase = lds_byte_off(&smem[0]) + (unsigned)(lane * 16);
    for (int c = wid; c < CCH; c += 4) {            // wave w: channels w, w+4, ...
        unsigned lo = lbase + (unsigned)(c * WW * 4);
        unsigned long long ga =
            (unsigned long long)(row0 + (size_t)c * (HH * WW)) + (unsigned)(lane * 16);
        asm volatile("global_load_async_to_lds_b128 %0, %1, off"
                     :: "v"(lo), "v"(ga) : "memory");
    }
    asm volatile("s_wait_asynccnt 0x0" ::: "memory");
    __syncthreads();                                 // publish tile across waves

    // ---- per-cell truth ----
    const int cell = (b * HH + y) * WW + x;
    const int am = amap[cell];
    float4 tb = make_float4(0.f, 0.f, 0.f, 0.f);
    float tcls = 0.f;
    if (am >= 0) { tb = bmap[cell]; tcls = cmap[cell]; }

    // hoist anchor constants (uniform scalars) out of the hot loop
    float awv[AA], ahv[AA];
#pragma unroll
    for (int a = 0; a < AA; ++a) {
        awv[a] = anchors[a * 2 + 0] / RESV;
        ahv[a] = anchors[a * 2 + 1] / RESV;
    }

    float xy_s = 0.f, wh_s = 0.f, obj_s = 0.f, noobj_s = 0.f, cls_s = 0.f;
    float nob = 0.f, nnob = 0.f;
    const float gx = (float)x, gy = (float)y;

#pragma unroll
    for (int a = 0; a < AA; ++a) {
        float o[10];
#pragma unroll
        for (int c = 0; c < 10; ++c) o[c] = smem[(a * 10 + c) * WW + x];

        float px = 1.f / (1.f + expf(-o[0])) + gx;
        float py = 1.f / (1.f + expf(-o[1])) + gy;
        float pw = sqrtf(expf(o[2]) * awv[a]);
        float ph = sqrtf(expf(o[3]) * ahv[a]);
        float pc = 1.f / (1.f + expf(-o[4]));

        if (a == am) {
            float dx = px - tb.x, dy = py - tb.y;
            xy_s += dx * dx + dy * dy;
            float dw = pw - sqrtf(tb.z), dh = ph - sqrtf(tb.w);
            wh_s += dw * dw + dh * dh;
            // elem IoU, replicating the reference's pixel-vs-grid mismatch:
            // pred scaled by res, true kept as box/res.
            float b1cx = px * RESV, b1cy = py * RESV, b1w = pw * RESV, b1h = ph * RESV;
            float tl1x = b1cx - b1w * 0.5f, br1x = b1cx + b1w * 0.5f;
            float tl1y = b1cy - b1h * 0.5f, br1y = b1cy + b1h * 0.5f;
            float tl2x = tb.x - tb.z * 0.5f, br2x = tb.x + tb.z * 0.5f;
            float tl2y = tb.y - tb.w * 0.5f, br2y = tb.y + tb.w * 0.5f;
            float sx = fmaxf(fminf(br1x, br2x) - fmaxf(tl1x, tl2x), 0.f);
            float sy = fmaxf(fminf(br1y, br2y) - fmaxf(tl1y, tl2y), 0.f);
            float inter = sx * sy;
            float s1 = fmaxf(br1x - tl1x, 1e-6f) * fmaxf(br1y - tl1y, 1e-6f);
            float s2 = fmaxf(br2x - tl2x, 1e-6f) * fmaxf(br2y - tl2y, 1e-6f);
            float iou = inter / (s1 + s2 - inter);
            float dc = pc - iou;
            obj_s += dc * dc;
            // class NLL via log-sum-exp (== -log_softmax[target])
            float m = o[5];
#pragma unroll
            for (int k = 6; k < 10; ++k) m = fmaxf(m, o[k]);
            float se = 0.f;
#pragma unroll
            for (int k = 5; k < 10; ++k) se += expf(o[k] - m);
            int tc = (int)tcls; tc = tc < 0 ? 0 : (tc > NCLS - 1 ? NCLS - 1 : tc);
            cls_s += (logf(se) + m) - o[5 + tc];
            nob += 1.f;
        } else {
            // true box is all-zero here => intersection == 0 => iou == 0 exactly
            noobj_s += pc * pc;
            nnob += 1.f;
        }
    }

    // ---- deterministic block reduction: wave32 butterfly + LDS ----
    float vals[7] = {xy_s, wh_s, obj_s, noobj_s, cls_s, nob, nnob};
#pragma unroll
    for (int k = 0; k < 7; ++k) {
        float v = vals[k];
#pragma unroll
        for (int off = 16; off > 0; off >>= 1) v += __shfl_xor(v, off, 32);
        if (lane == 0) red[wid * 8 + k] = v;
    }
    __syncthreads();
    if (threadIdx.x == 0) {
#pragma unroll
        for (int k = 0; k < 7; ++k)
            partials[(size_t)bid * 8 + k] = red[k] + red[8 + k] + red[16 + k] + red[24 + k];
    }
}

// ---------------- kernel 4: deterministic final reduce + loss -----------
__global__ void __launch_bounds__(256)
k_final(const float* __restrict__ partials, int nblocks, float* __restrict__ outp) {
    __shared__ float acc[256 * 8];
    const int t = threadIdx.x;
    float s[7] = {0.f, 0.f, 0.f, 0.f, 0.f, 0.f, 0.f};
    for (int i = t; i < nblocks; i += 256)
#pragma unroll
        for (int k = 0; k < 7; ++k) s[k] += partials[(size_t)i * 8 + k];
#pragma unroll
    for (int k = 0; k < 7; ++k) acc[t * 8 + k] = s[k];
    __syncthreads();
    for (int stride = 128; stride > 0; stride >>= 1) {
        if (t < stride)
#pragma unroll
            for (int k = 0; k < 7; ++k) acc[t * 8 + k] += acc[(t + stride) * 8 + k];
        __syncthreads();
    }
    if (t == 0) {
        float n_obj = acc[5], n_noobj = acc[6];
        float xy_loss = acc[0] / (2.f * n_obj);
        float wh_loss = acc[1] / (2.f * n_obj);
        float obj_loss = acc[2] / n_obj;
        float noobj_loss = acc[3] / n_noobj;
        float class_loss = acc[4] / n_obj;
        outp[0] = 5.0f * (xy_loss + wh_loss) + 1.0f * obj_loss
                + 0.5f * noobj_loss + 1.0f * class_loss;
    }
}

// ---------------- launcher ----------------------------------------------
extern "C" void kernel_launch(void* const* d_in, const int* in_sizes, int n_in,
                              void* d_out, int out_size, void* d_ws, size_t ws_size,
                              hipStream_t stream) {
    const float* out_t   = (const float*)d_in[0];   // [B,50,128,128]
    const float* labels  = (const float*)d_in[1];   // [N,6]
    const float* anchors = (const float*)d_in[2];   // [5,2]
    const int Bimgs = in_sizes[0] / (CCH * HH * WW);   // 32
    const int N     = in_sizes[1] / 6;                 // 640
    const int ncell = Bimgs * HH * WW;                 // 524288
    const int nblk  = Bimgs * HH;                      // 4096

    // workspace layout (all reinitialized every call):
    //   amap: int[ncell], bmap: float4[ncell], cmap: float[ncell],
    //   partials: float[nblk*8]   => ~12.7 MB total
    char* ws = (char*)d_ws;
    int*    amap = (int*)ws;
    float4* bmap = (float4*)(ws + (size_t)ncell * 4);
    float*  cmap = (float*)(ws + (size_t)ncell * 4 + (size_t)ncell * 16);
    float*  partials = (float*)(ws + (size_t)ncell * 4 + (size_t)ncell * 16
                                   + (size_t)ncell * 4);

    k_init<<<(ncell + 255) / 256, 256, 0, stream>>>(amap, ncell);
    k_scatter<<<(N + 63) / 64, 64, 0, stream>>>(labels, anchors, N, amap, bmap, cmap);
    k_main<<<nblk, 128, 0, stream>>>(out_t, anchors, amap, bmap, cmap, partials);
    k_final<<<1, 256, 0, stream>>>(partials, nblk, (float*)d_out);
}